// EnhancedEntityNBFNet_27015344291946
// MI455X (gfx1250) — compile-verified
//
#include <hip/hip_runtime.h>

// Problem constants (from reference): B=2, K=33, N=50000, E=800000, R=200, D=64, L=3
static constexpr int DIM  = 64;
static constexpr int TWOD = 128;

typedef __attribute__((ext_vector_type(2))) float v2f;
typedef __attribute__((ext_vector_type(8))) float v8f;

// ---------------------------------------------------------------------------
// x[b, h0] = query = rel[b, r0]   (rest of x already zeroed by memset)
// ---------------------------------------------------------------------------
__global__ void init_x_kernel(float* __restrict__ x, const float* __restrict__ rel,
                              const int* __restrict__ batch, int N, int Kq, int Rr) {
    int b = blockIdx.x;
    int t = threadIdx.x;
    int h0 = batch[b * Kq * 3 + 0];
    int r0 = batch[b * Kq * 3 + 2];
    x[((long)b * N + h0) * DIM + t] = rel[((long)b * Rr + r0) * DIM + t];
}

// ---------------------------------------------------------------------------
// Edge scatter: agg[b, dst] += x[b, src] * rel[b, etype]
// 16 lanes per (edge, batch), float4 per lane -> 64 floats, coalesced.
// atomicAdd(float) lowers to global_atomic_add_f32 (no-return, STOREcnt).
// Speculative prefetch of a future edge's source row (global_prefetch_b8)
// warms L2 for the gather stream.
// ---------------------------------------------------------------------------
__global__ __launch_bounds__(256)
void scatter_kernel(const float* __restrict__ x, const float* __restrict__ rel,
                    float* __restrict__ agg,
                    const int* __restrict__ ei, const int* __restrict__ et,
                    int E, int N, int Rr) {
    int b    = blockIdx.y;
    int e    = blockIdx.x * 16 + (threadIdx.x >> 4);
    int lane = threadIdx.x & 15;
    if (e >= E) return;

    // prefetch the gather row of an edge well ahead of this one
    int e2 = e + 4096;
    if (e2 < E) {
        int s2 = ei[e2];
        __builtin_prefetch(x + ((long)b * N + s2) * DIM + lane * 4, 0, 0);
    }

    int s = ei[e];
    int d = ei[E + e];
    int r = et[e];
    const float4 xv = *(const float4*)(x   + ((long)b * N  + s) * DIM + lane * 4);
    const float4 rv = *(const float4*)(rel + ((long)b * Rr + r) * DIM + lane * 4);
    float* dst = agg + ((long)b * N + d) * DIM + lane * 4;
    atomicAdd(dst + 0, xv.x * rv.x);
    atomicAdd(dst + 1, xv.y * rv.y);
    atomicAdd(dst + 2, xv.z * rv.z);
    atomicAdd(dst + 3, xv.w * rv.w);
}

// ---------------------------------------------------------------------------
// Node update: upd = LN(concat(agg + boundary, x) @ W + bias); x = relu(upd)+x
// Block = 128 threads (4 waves), 32 nodes per block.
// Wave w: M-tile = w>>1 (16 rows), N-cols = (w&1)*32 .. +31 (2 WMMA N-tiles).
// fp32 WMMA: V_WMMA_F32_16X16X4_F32, K looped 128/4 = 32 steps.
// W staged TRANSPOSED in LDS ([col][k], stride 132) so every A and B fragment
// is a single aligned ds_load_b64 with no packing movs and no bank conflicts.
// ---------------------------------------------------------------------------
__global__ __launch_bounds__(128)
void update_kernel(const float* __restrict__ agg, float* __restrict__ x,
                   const float* __restrict__ rel,
                   const float* __restrict__ W, const float* __restrict__ lb,
                   const float* __restrict__ lng, const float* __restrict__ lnb,
                   const int* __restrict__ batch, int N, int Kq, int Rr) {
    __shared__ float sW[DIM * 132];   // W^T: sW[col*132 + k], k in [0,128)
    __shared__ float sA[32 * 132];    // A rows: [agg+boundary | x], stride 132
    __shared__ float red_s[32 * 4];
    __shared__ float red_q[32 * 4];

    const int tid   = threadIdx.x;
    const int b     = blockIdx.y;
    const int node0 = blockIdx.x * 32;
    const int h0 = batch[b * Kq * 3 + 0];
    const int r0 = batch[b * Kq * 3 + 2];

    // ---- stage W (128x64) into LDS transposed; global reads float4-coalesced ----
    for (int i = 0; i < 16; ++i) {
        int idx = (tid + 128 * i) * 4;          // 8192 floats
        int k   = idx >> 6;
        int col = idx & 63;
        float4 wv = *(const float4*)(W + k * DIM + col);
        sW[(col + 0) * 132 + k] = wv.x;
        sW[(col + 1) * 132 + k] = wv.y;
        sW[(col + 2) * 132 + k] = wv.z;
        sW[(col + 3) * 132 + k] = wv.w;
    }
    // ---- stage A (32 x 128): cols 0..63 = agg + boundary, 64..127 = x ----
    for (int i = 0; i < 8; ++i) {
        int idx = (tid + 128 * i) * 4;          // 4096 floats
        int row = idx >> 7;
        int col = idx & 127;
        int node = node0 + row;
        float4 av = make_float4(0.f, 0.f, 0.f, 0.f);
        if (node < N) {
            if (col < 64) {
                av = *(const float4*)(agg + ((long)b * N + node) * DIM + col);
                if (node == h0) {
                    float4 qv = *(const float4*)(rel + ((long)b * Rr + r0) * DIM + col);
                    av.x += qv.x; av.y += qv.y; av.z += qv.z; av.w += qv.w;
                }
            } else {
                av = *(const float4*)(x + ((long)b * N + node) * DIM + (col - 64));
            }
        }
        float* p = &sA[row * 132 + col];
        p[0] = av.x; p[1] = av.y; p[2] = av.z; p[3] = av.w;
    }
    __syncthreads();

    // ---- WMMA phase ----
    const int w   = tid >> 5;
    const int tl  = tid & 31;
    const int row = tl & 15;     // M (A) / N (B,C,D) within tile
    const int hi  = tl >> 4;     // K half (A,B) / M+8 half (C,D)
    const int mr  = (w >> 1) * 16;
    const int cb  = (w & 1) * 32;

    const float* pA  = &sA[(mr + row) * 132];
    const float* pB0 = &sW[(cb + row) * 132];
    const float* pB1 = &sW[(cb + 16 + row) * 132];

    v8f acc0 = {};
    v8f acc1 = {};
    for (int kk = 0; kk < 32; ++kk) {
        int k0 = kk * 4 + 2 * hi;                 // even -> 8B-aligned b64 loads
        v2f a  = *(const v2f*)(pA  + k0);
        v2f b0 = *(const v2f*)(pB0 + k0);
        v2f b1 = *(const v2f*)(pB1 + k0);
        acc0 = __builtin_amdgcn_wmma_f32_16x16x4_f32(false, a, false, b0,
                                                     (short)0, acc0, false, false);
        acc1 = __builtin_amdgcn_wmma_f32_16x16x4_f32(false, a, false, b1,
                                                     (short)0, acc1, false, false);
    }
    __syncthreads();   // all reads of sA cols 0..63 done before overwrite

    // ---- write D (16x32 per wave) into sA cols 0..63 ----
    for (int vg = 0; vg < 8; ++vg) {
        int m = mr + vg + 8 * hi;
        sA[m * 132 + cb + row]      = acc0[vg];
        sA[m * 132 + cb + 16 + row] = acc1[vg];
    }
    __syncthreads();

    // ---- LayerNorm + ReLU + residual (thread -> row tid&31, 16-col segment) ----
    {
        int rr  = tid & 31;
        int seg = tid >> 5;
        float vloc[16];
        float s = 0.f, q = 0.f;
        for (int j = 0; j < 16; ++j) {
            int c = seg * 16 + j;
            float v = sA[rr * 132 + c] + lb[c];
            vloc[j] = v;
            s += v; q += v * v;
        }
        red_s[rr * 4 + seg] = s;
        red_q[rr * 4 + seg] = q;
        __syncthreads();
        float ts = red_s[rr*4+0] + red_s[rr*4+1] + red_s[rr*4+2] + red_s[rr*4+3];
        float tq = red_q[rr*4+0] + red_q[rr*4+1] + red_q[rr*4+2] + red_q[rr*4+3];
        float mu  = ts * (1.f / 64.f);
        float var = tq * (1.f / 64.f) - mu * mu;
        float inv = rsqrtf(var + 1e-5f);
        for (int j = 0; j < 16; ++j) {
            int c = seg * 16 + j;
            float v = (vloc[j] - mu) * inv * lng[c] + lnb[c];
            v = fmaxf(v, 0.f);
            v += sA[rr * 132 + 64 + c];   // residual: old x (cols 64..127 intact)
            sA[rr * 132 + c] = v;
        }
    }
    __syncthreads();

    // ---- coalesced float4 store of new x (32 x 64) ----
    for (int i = 0; i < 4; ++i) {
        int idx = (tid + 128 * i) * 4;    // 2048 floats
        int rr  = idx >> 6;
        int col = idx & 63;
        int node = node0 + rr;
        if (node < N) {
            float* p = &sA[rr * 132 + col];
            float4 ov; ov.x = p[0]; ov.y = p[1]; ov.z = p[2]; ov.w = p[3];
            *(float4*)(x + ((long)b * N + node) * DIM + col) = ov;
        }
    }
}

// ---------------------------------------------------------------------------
// Final MLP score at t_index: relu(concat(x[b,t], query) @ w1 + b1) @ w2 + b2
// One block (64 threads) per (b, k); 66 blocks total.
// ---------------------------------------------------------------------------
__global__ __launch_bounds__(64)
void score_kernel(const float* __restrict__ x, const float* __restrict__ rel,
                  const float* __restrict__ w1, const float* __restrict__ b1,
                  const float* __restrict__ w2, const float* __restrict__ b2,
                  const int* __restrict__ batch, float* __restrict__ out,
                  int N, int Kq, int Rr) {
    __shared__ float feat[TWOD];
    __shared__ float part[DIM];
    int b = blockIdx.x / Kq;
    int k = blockIdx.x % Kq;
    int j = threadIdx.x;
    int t  = batch[(b * Kq + k) * 3 + 1];
    int r0 = batch[b * Kq * 3 + 2];
    feat[j]      = x[((long)b * N + t) * DIM + j];
    feat[j + 64] = rel[((long)b * Rr + r0) * DIM + j];
    __syncthreads();
    float h = b1[j];
    for (int i = 0; i < TWOD; ++i) h += feat[i] * w1[i * DIM + j];
    h = fmaxf(h, 0.f);
    part[j] = h * w2[j];
    __syncthreads();
    for (int sft = 32; sft > 0; sft >>= 1) {
        if (j < sft) part[j] += part[j + sft];
        __syncthreads();
    }
    if (j == 0) out[b * Kq + k] = part[0] + b2[0];
}

// ---------------------------------------------------------------------------
extern "C" void kernel_launch(void* const* d_in, const int* in_sizes, int n_in,
                              void* d_out, int out_size, void* d_ws, size_t ws_size,
                              hipStream_t stream) {
    const float* rel      = (const float*)d_in[0];
    const float* layer_w  = (const float*)d_in[1];
    const float* layer_b  = (const float*)d_in[2];
    const float* ln_g     = (const float*)d_in[3];
    const float* ln_b     = (const float*)d_in[4];
    const float* mlp_w1   = (const float*)d_in[5];
    const float* mlp_b1   = (const float*)d_in[6];
    const float* mlp_w2   = (const float*)d_in[7];
    const float* mlp_b2   = (const float*)d_in[8];
    const int*   batch    = (const int*)d_in[9];
    const int*   edge_idx = (const int*)d_in[10];
    const int*   edge_typ = (const int*)d_in[11];

    const int B = 2, Kq = 33, N = 50000, E = 800000, Rr = 200, L = 3;

    float* x   = (float*)d_ws;                      // B*N*64 floats = 25.6 MB
    float* agg = x + (size_t)B * N * DIM;           // B*N*64 floats = 25.6 MB
    const size_t state_bytes = (size_t)B * N * DIM * sizeof(float);

    hipMemsetAsync(x, 0, state_bytes, stream);
    init_x_kernel<<<B, DIM, 0, stream>>>(x, rel, batch, N, Kq, Rr);

    for (int l = 0; l < L; ++l) {
        hipMemsetAsync(agg, 0, state_bytes, stream);
        dim3 sg((E + 15) / 16, B);
        scatter_kernel<<<sg, 256, 0, stream>>>(x, rel, agg, edge_idx, edge_typ, E, N, Rr);
        dim3 ug((N + 31) / 32, B);
        update_kernel<<<ug, 128, 0, stream>>>(agg, x, rel,
            layer_w + (size_t)l * TWOD * DIM, layer_b + (size_t)l * DIM,
            ln_g + (size_t)l * DIM, ln_b + (size_t)l * DIM,
            batch, N, Kq, Rr);
    }

    score_kernel<<<B * Kq, DIM, 0, stream>>>(x, rel, mlp_w1, mlp_b1, mlp_w2, mlp_b2,
                                             batch, (float*)d_out, N, Kq, Rr);
}